// HarrisNet_68693706932902
// MI455X (gfx1250) — compile-verified
//
#include <hip/hip_runtime.h>
#include <stdint.h>

#define IMG_N 4096
#define IMG_SZ (IMG_N * IMG_N)
#define ALPHA_C 0.05f
#define NB 586

// Normalized separable Gaussian taps (sigma=5, ksize=7) as literals
#define G0 0.12895594f
#define G1 0.14251791f
#define G2 0.15133123f
#define G3 0.15438852f

typedef float v2f __attribute__((ext_vector_type(2)));
typedef float v8f __attribute__((ext_vector_type(8)));
typedef int avi4 __attribute__((vector_size(16)));  // matches builtin param

#if __has_builtin(__builtin_amdgcn_global_load_async_to_lds_b128)
#define HAS_ASYNC_LDS 1
#endif

__device__ __forceinline__ void wait_async_lds() {
#if __has_builtin(__builtin_amdgcn_s_wait_asynccnt)
  __builtin_amdgcn_s_wait_asynccnt(0);
#else
  asm volatile("s_wait_asynccnt 0x0" ::: "memory");
#endif
}

// Band coefficient gn[d] for d in [0,6], 0 outside; pure VALU select chain.
__device__ __forceinline__ float bandcoef(int d) {
  int e = (d > 3) ? (6 - d) : d;  // fold symmetric taps; <0 => zero
  float v = 0.0f;
  v = (e == 0) ? G0 : v;
  v = (e == 1) ? G1 : v;
  v = (e == 2) ? G2 : v;
  v = (e == 3) ? G3 : v;
  return v;
}

// ---------------- fused Sobel + products + separable Gaussian + Harris R ----
#define TILE_W 64
#define TILE_H 16
#define XT_R 24
#define XT_C 72
#define XT_S 76  // row stride: 76*4 bytes = 16B-aligned rows for b128 async
#define P_R 22
#define P_C 70
#define P_S 71
#define H_R 22
#define H_S 65

__global__ __launch_bounds__(128) void harris_fused(
    const float* __restrict__ x, float* __restrict__ R) {
  __shared__ float xt[XT_R * XT_S];
  __shared__ float P[3][P_R * P_S];
  __shared__ float Hs[3][H_R * H_S];

  const int tid = threadIdx.x;
  const int col0 = blockIdx.x * TILE_W;
  const int row0 = blockIdx.y * TILE_H;

  // Stage 1: load x tile with halo 4 (zero-padded outside image)
  const bool interior = (row0 >= 4) && (row0 + XT_R - 4 <= IMG_N) &&
                        (col0 >= 4) && (col0 + XT_C - 4 <= IMG_N);
  if (interior) {
#ifdef HAS_ASYNC_LDS
    // 24 rows x 18 float4 each, 16B-aligned on both ends.
    for (int t = tid; t < XT_R * (XT_C / 4); t += 128) {
      int r = t / (XT_C / 4), c4 = t % (XT_C / 4);
      const float* src = x + (size_t)(row0 + r - 4) * IMG_N + (col0 - 4 + c4 * 4);
      float* dst = &xt[r * XT_S + c4 * 4];
      __builtin_amdgcn_global_load_async_to_lds_b128(
          (__attribute__((address_space(1))) avi4*)src,
          (__attribute__((address_space(3))) avi4*)dst, 0, 0);
    }
    wait_async_lds();
#else
    for (int t = tid; t < XT_R * XT_C; t += 128) {
      int r = t / XT_C, c = t % XT_C;
      xt[r * XT_S + c] = x[(size_t)(row0 + r - 4) * IMG_N + (col0 + c - 4)];
    }
#endif
  } else {
    for (int t = tid; t < XT_R * XT_C; t += 128) {
      int r = t / XT_C, c = t % XT_C;
      int gr = row0 + r - 4, gc = col0 + c - 4;
      float v = 0.0f;
      if (gr >= 0 && gr < IMG_N && gc >= 0 && gc < IMG_N)
        v = x[gr * IMG_N + gc];
      xt[r * XT_S + c] = v;
    }
  }
  __syncthreads();

  // Stage 2: Sobel (cross-correlation) + products on rows -3..18, cols -3..66.
  // Products are zero outside [0,4096) (Gaussian conv zero-pads products).
  for (int t = tid; t < P_R * P_C; t += 128) {
    int r = t / P_C, c = t % P_C;
    int gr = row0 + r - 3, gc = col0 + c - 3;
    bool inimg = (gr >= 0 && gr < IMG_N && gc >= 0 && gc < IMG_N);
    const float* b0 = &xt[(r)*XT_S + c];
    const float* b1 = &xt[(r + 1) * XT_S + c];
    const float* b2 = &xt[(r + 2) * XT_S + c];
    float ix = (b0[2] - b0[0]) + 2.0f * (b1[2] - b1[0]) + (b2[2] - b2[0]);
    float iy = (b2[0] + 2.0f * b2[1] + b2[2]) - (b0[0] + 2.0f * b0[1] + b0[2]);
    if (!inimg) { ix = 0.0f; iy = 0.0f; }
    P[0][r * P_S + c] = ix * ix;
    P[1][r * P_S + c] = iy * iy;
    P[2][r * P_S + c] = ix * iy;
  }
  __syncthreads();

  // Stage 3: horizontal 7-tap blur: 22 rows x 64 cols x 3 channels
  for (int t = tid; t < H_R * TILE_W; t += 128) {
    int r = t / TILE_W, c = t % TILE_W;
#pragma unroll
    for (int ch = 0; ch < 3; ++ch) {
      const float* p = &P[ch][r * P_S + c];
      float s = G0 * (p[0] + p[6]);
      s += G1 * (p[1] + p[5]);
      s += G2 * (p[2] + p[4]);
      s += G3 * p[3];
      Hs[ch][r * H_S + c] = s;
    }
  }
  __syncthreads();

  // Stage 4: vertical 7-tap blur as banded 16x22 @ 22x16 matmul via chained
  // V_WMMA_F32_16X16X4_F32 (K padded to 24 with zero band coefficients).
  // One 16x16 subtile per wave; uniform control flow => EXEC all ones.
  const int wave = tid >> 5;
  const int lane = tid & 31;
  const int m = lane & 15;   // M row (A) / N col (B,C)
  const int hi = lane >> 4;  // lane-half select per ISA VGPR layout
  const int n0 = wave * 16;

  // A matrix (channel-invariant): VGPR0 = K {0,2}, VGPR1 = K {1,3} per half.
  v2f Aa[6];
#pragma unroll
  for (int kc = 0; kc < 6; ++kc) {
    const int kb = kc * 4;
    Aa[kc].x = bandcoef(kb + 2 * hi - m);
    Aa[kc].y = bandcoef(kb + 1 + 2 * hi - m);
  }

  v8f acc[3];
#pragma unroll
  for (int ch = 0; ch < 3; ++ch) {
    v8f c = {};
#pragma unroll
    for (int kc = 0; kc < 6; ++kc) {
      const int kb = kc * 4;
      // B (4x16 f32): rows beyond 21 carry zero A-coeffs -> clamp row index
      int r0k = kb + 2 * hi;      if (r0k > 21) r0k = 21;
      int r1k = kb + 1 + 2 * hi;  if (r1k > 21) r1k = 21;
      v2f b;
      b.x = Hs[ch][r0k * H_S + n0 + m];
      b.y = Hs[ch][r1k * H_S + n0 + m];
      c = __builtin_amdgcn_wmma_f32_16x16x4_f32(
          /*neg_a=*/false, Aa[kc], /*neg_b=*/false, b,
          /*c_mod=*/(short)0, c, /*reuse_a=*/false, /*reuse_b=*/false);
    }
    acc[ch] = c;
  }

  // Harris response from accumulators; C/D layout: VGPR v -> M = v + 8*hi
#pragma unroll
  for (int v = 0; v < 8; ++v) {
    float sx = acc[0][v], sy = acc[1][v], sxy = acc[2][v];
    float tr = sx + sy;
    float r = sx * sy - sxy * sxy - ALPHA_C * tr * tr;
    int gr = row0 + v + 8 * hi;
    int gc = col0 + n0 + m;
    R[gr * IMG_N + gc] = r;
  }
}

// ---------------- median via 4-pass radix select --------------------------
__device__ __forceinline__ uint32_t f2ord(float f) {
  uint32_t b = __float_as_uint(f);
  return (b & 0x80000000u) ? ~b : (b | 0x80000000u);
}
__device__ __forceinline__ float ord2f(uint32_t u) {
  uint32_t b = (u & 0x80000000u) ? (u & 0x7fffffffu) : ~u;
  return __uint_as_float(b);
}

__global__ void med_init(uint32_t* hist, uint32_t* state) {
  int t = threadIdx.x;
  hist[t] = 0u;
  if (t == 0) {
    state[0] = 0u;        // selected high bits so far
    state[1] = 8388607u;  // target rank = (N-1)//2
  }
}

__global__ __launch_bounds__(256) void med_hist(
    const float* __restrict__ R, uint32_t* __restrict__ hist,
    const uint32_t* __restrict__ state, int shift) {
  __shared__ uint32_t lh[256];
  lh[threadIdx.x] = 0u;
  __syncthreads();
  const uint32_t prefix = state[0];
  const uint32_t hs = (uint32_t)(shift + 8);
  const int stride = gridDim.x * blockDim.x;
  for (int i = blockIdx.x * blockDim.x + threadIdx.x; i < IMG_SZ; i += stride) {
    __builtin_prefetch(&R[i + stride], 0, 0);
    uint32_t u = f2ord(R[i]);
    bool ok = (hs >= 32u) ? true : ((u >> hs) == prefix);
    if (ok) atomicAdd(&lh[(u >> shift) & 255u], 1u);
  }
  __syncthreads();
  uint32_t v = lh[threadIdx.x];
  if (v) atomicAdd(&hist[threadIdx.x], v);
}

__global__ void med_select(uint32_t* hist, uint32_t* state, float* med,
                           int last) {
  __shared__ uint32_t h[256];
  int t = threadIdx.x;
  h[t] = hist[t];
  __syncthreads();
  if (t == 0) {
    uint32_t rank = state[1];
    uint32_t run = 0, sel = 255;
    for (int i = 0; i < 256; ++i) {
      uint32_t nr = run + h[i];
      if (rank < nr) { sel = (uint32_t)i; break; }
      run = nr;
    }
    state[0] = (state[0] << 8) | sel;
    state[1] = rank - run;
    if (last) *med = ord2f(state[0]);
  }
  __syncthreads();
  hist[t] = 0u;  // ready for next pass
}

// ---------------- 7x7 stride-7 block max (pad 3) --------------------------
__global__ __launch_bounds__(256) void block_max(
    const float* __restrict__ R, const float* __restrict__ med,
    float* __restrict__ bm) {
  int t = blockIdx.x * blockDim.x + threadIdx.x;
  if (t >= NB * NB) return;
  int bi = t / NB, bj = t % NB;
  float mv = *med;
  float m = -INFINITY;
  int rbase = bi * 7 - 3, cbase = bj * 7 - 3;
  for (int dr = 0; dr < 7; ++dr) {
    int r = rbase + dr;
    if (r < 0 || r >= IMG_N) continue;
    for (int dc = 0; dc < 7; ++dc) {
      int c = cbase + dc;
      if (c < 0 || c >= IMG_N) continue;
      float v = R[r * IMG_N + c];
      v = (v < mv) ? 0.0f : v;
      m = fmaxf(m, v);
    }
  }
  bm[t] = m;
}

// ---------------- threshold + keep-equal-to-blockmax ----------------------
__global__ __launch_bounds__(256) void finalize(
    const float* __restrict__ R, const float* __restrict__ med,
    const float* __restrict__ bm, float* __restrict__ out) {
  int i = blockIdx.x * blockDim.x + threadIdx.x;
  if (i >= IMG_SZ) return;
  int r = i >> 12, c = i & 4095;
  float mv = *med;
  float v = R[i];
  v = (v < mv) ? 0.0f : v;
  float m = bm[((r + 3) / 7) * NB + ((c + 3) / 7)];
  out[i] = (v == m) ? v : 0.0f;
}

// ---------------- launcher -------------------------------------------------
extern "C" void kernel_launch(void* const* d_in, const int* in_sizes, int n_in,
                              void* d_out, int out_size, void* d_ws,
                              size_t ws_size, hipStream_t stream) {
  (void)in_sizes; (void)n_in; (void)out_size; (void)ws_size;
  const float* x = (const float*)d_in[0];  // weights are baked-in constants

  char* base = (char*)d_ws;
  float* R = (float*)base;                                  // 64 MB
  uint32_t* hist = (uint32_t*)(base + (size_t)IMG_SZ * 4);  // 1 KB
  uint32_t* state = hist + 256;                             // 8 B
  float* med = (float*)(state + 2);                         // 4 B
  float* bm = (float*)(base + (size_t)IMG_SZ * 4 + 4096);   // ~1.4 MB
  float* out = (float*)d_out;

  dim3 gridF(IMG_N / TILE_W, IMG_N / TILE_H);
  harris_fused<<<gridF, 128, 0, stream>>>(x, R);

  med_init<<<1, 256, 0, stream>>>(hist, state);
  for (int p = 0; p < 4; ++p) {
    int shift = 24 - 8 * p;
    med_hist<<<2048, 256, 0, stream>>>(R, hist, state, shift);
    med_select<<<1, 256, 0, stream>>>(hist, state, med, p == 3 ? 1 : 0);
  }

  block_max<<<(NB * NB + 255) / 256, 256, 0, stream>>>(R, med, bm);
  finalize<<<(IMG_SZ + 255) / 256, 256, 0, stream>>>(R, med, bm, out);
}